// MultiHeadAttention_80290118631470
// MI455X (gfx1250) — compile-verified
//
#include <hip/hip_runtime.h>
#include <hip/hip_bf16.h>

// MI455X (gfx1250) fp32 multi-head attention using V_WMMA_F32_16X16X4_F32
// plus Tensor Data Mover (tensor_load_to_lds) for K/V tile staging.
// Shapes: B=8, NQ=NK=1024, D=512, H=8, DK=DV=64.
// d_out = [ out: 8*1024*512 floats | unnorm_attn: 8*8*1024*1024 floats ]

typedef __attribute__((ext_vector_type(2))) float v2f;
typedef __attribute__((ext_vector_type(8))) float v8f;
typedef __attribute__((ext_vector_type(4))) unsigned int u32x4;
typedef __attribute__((ext_vector_type(8))) int i32x8;

#define NEG_INF (-__builtin_inff())

__device__ __forceinline__ v8f wmma4(v2f a, v2f b, v8f c) {
  // D = A(16x4,f32) * B(4x16,f32) + C(16x16,f32)
  return __builtin_amdgcn_wmma_f32_16x16x4_f32(
      /*neg_a=*/false, a, /*neg_b=*/false, b,
      /*c_mod=*/(short)0, c, /*reuse_a=*/false, /*reuse_b=*/false);
}

// ---------------------------------------------------------------------------
// TDM: DMA a [rows x 64] f32 tile (global row stride 64 floats) into LDS with
// a 4-dword pad after every 64-dword row (LDS row stride 68 floats).
// D# layout per CDNA5 ISA ch.8; 2-operand form (VADDR2/3 = NULL, <=2D tensor).
// ---------------------------------------------------------------------------
__device__ __forceinline__ void tdm_load_tile64_f32(const float* gsrc,
                                                    unsigned lds_off,
                                                    unsigned rows) {
  const unsigned long long ga = (unsigned long long)(uintptr_t)gsrc;
  u32x4 g0;
  g0.x = 1u;                                   // count=1 (valid user descriptor)
  g0.y = lds_off;                              // lds_addr (bytes)
  g0.z = (unsigned)ga;                         // global_addr[31:0]
  g0.w = (unsigned)(ga >> 32) | (2u << 30);    // global_addr[56:32] | type=2
  i32x8 g1;
  g1[0] = (int)((2u << 16)      // data_size = 2 -> 4 bytes
              | (1u << 20)      // pad_enable
              | (5u << 22)      // pad_interval = 5 -> every 64 dwords
              | (3u << 25));    // pad_amount  = 3 -> 4 dwords
  g1[1] = (int)(64u << 16);     // tensor_dim0 = 64 (atomic_barrier_addr = 0)
  g1[2] = (int)(1024u << 16);   // tensor_dim1 = 1024 (tensor_dim0 hi = 0)
  g1[3] = (int)(64u << 16);     // tile_dim0 = 64 (tensor_dim1 hi = 0)
  g1[4] = (int)rows;            // tile_dim1 = rows, tile_dim2 = 0 (2D)
  g1[5] = 64;                   // tensor_dim0_stride = 64 elements
  g1[6] = 0;                    // stride hi / tensor_dim1_stride lo
  g1[7] = 0;                    // tensor_dim1_stride hi (unused, 2D tile)
  asm volatile("tensor_load_to_lds %0, %1" :: "s"(g0), "s"(g1) : "memory");
}

// ---------------------------------------------------------------------------
// Tiled GEMM: Y = A[M,512] @ W[512,512] + bias.  64x64 C-tile per block.
// scatter==1: write to head-major [(b*8+h)*1024+nq]*64+d (for QKV proj).
// scatter==0: plain row-major [M,512] (for output projection).
// ---------------------------------------------------------------------------
__global__ __launch_bounds__(256)
void gemm512_wmma_kernel(const float* __restrict__ A,
                         const float* __restrict__ W,
                         const float* __restrict__ bias,
                         float* __restrict__ Y,
                         int scatter)
{
  __shared__ float Xs[64][40];   // 64x32 tile of A, stride 40 (16B-aligned rows)
  __shared__ float Ws[32][68];   // 32x64 tile of W, stride 68

  const int tid  = threadIdx.x;
  const int lane = tid & 31;
  const int wid  = tid >> 5;     // 0..7
  const int tm   = wid & 3;      // C tile row (16 rows each)
  const int tn2  = wid >> 2;     // which 32-col half
  const int hi   = lane >> 4;    // lane half -> K pair select
  const int ln   = lane & 15;
  const int row0 = blockIdx.y * 64;
  const int col0 = blockIdx.x * 64;

  v8f acc0 = {}; v8f acc1 = {};

  for (int k0 = 0; k0 < 512; k0 += 32) {
    if (k0 + 32 < 512) {   // prefetch next A tile (global_prefetch_b8)
      __builtin_prefetch(A + (size_t)(row0 + (tid >> 2)) * 512 + k0 + 32, 0, 0);
    }
#pragma unroll
    for (int t = 0; t < 2; ++t) {            // A tile: 64x32 = 512 float4
      int i = tid + t * 256;
      int r = i >> 3, c4 = (i & 7) << 2;
      *(float4*)&Xs[r][c4] = *(const float4*)(A + (size_t)(row0 + r) * 512 + k0 + c4);
    }
#pragma unroll
    for (int t = 0; t < 2; ++t) {            // W tile: 32x64 = 512 float4
      int i = tid + t * 256;
      int r = i >> 4, c4 = (i & 15) << 2;
      *(float4*)&Ws[r][c4] = *(const float4*)(W + (size_t)(k0 + r) * 512 + col0 + c4);
    }
    __syncthreads();

#pragma unroll
    for (int kk = 0; kk < 32; kk += 4) {
      const int ka = kk + hi * 2;
      v2f a;  a.x  = Xs[tm * 16 + ln][ka];        a.y  = Xs[tm * 16 + ln][ka + 1];
      v2f b0; b0.x = Ws[ka][tn2 * 32 + ln];       b0.y = Ws[ka + 1][tn2 * 32 + ln];
      v2f b1; b1.x = Ws[ka][tn2 * 32 + 16 + ln];  b1.y = Ws[ka + 1][tn2 * 32 + 16 + ln];
      acc0 = wmma4(a, b0, acc0);
      acc1 = wmma4(a, b1, acc1);
    }
    __syncthreads();
  }

#pragma unroll
  for (int vr = 0; vr < 8; ++vr) {
    const int r   = row0 + tm * 16 + vr + hi * 8;   // C row per ISA layout
    const int c0g = col0 + tn2 * 32 + ln;
    const int c1g = c0g + 16;
    const float y0 = acc0[vr] + bias[c0g];
    const float y1 = acc1[vr] + bias[c1g];
    if (scatter) {
      const int b = r >> 10, nq = r & 1023;
      Y[(((size_t)(b * 8 + (c0g >> 6)) * 1024 + nq) << 6) + (c0g & 63)] = y0;
      Y[(((size_t)(b * 8 + (c1g >> 6)) * 1024 + nq) << 6) + (c1g & 63)] = y1;
    } else {
      Y[(size_t)r * 512 + c0g] = y0;
      Y[(size_t)r * 512 + c1g] = y1;
    }
  }
}

// ---------------------------------------------------------------------------
// Flash attention per (b,h): S = QK^T/8 (written unmasked to unnorm_attn),
// masked online softmax (true -inf; all-masked row -> exact 0), O = P V.
// 128 threads = 4 waves; each wave owns 16 query rows (intra-wave reductions).
// K/V tiles are staged into LDS by the Tensor Data Mover (wave 0 issues).
// ---------------------------------------------------------------------------
__global__ __launch_bounds__(128)
void attn_flash_wmma_kernel(const float* __restrict__ qh,   // [64][1024][64]
                            const float* __restrict__ kh,
                            const float* __restrict__ vh,
                            const int*   __restrict__ mask, // [8][1024][1024]
                            float* __restrict__ attn_out,   // [64][1024][1024]
                            float* __restrict__ o_cat)      // [8192][512]
{
  __shared__ float Qs[64][68];   // 64 q-rows x 64 dk
  __shared__ float Ks[32][68];   // 32 keys  x 64 dk  (TDM-filled, pad 4 dw/row)
  __shared__ float Vs[32][68];   // 32 keys  x 64 dv  (TDM-filled)
  __shared__ float Ps[64][36];   // 64 q-rows x 32 probs (C-frag -> A-frag relay)

  const int tid  = threadIdx.x;
  const int lane = tid & 31;
  const int wid  = tid >> 5;     // 0..3 -> q rows [16*wid, 16*wid+16)
  const int hi   = lane >> 4;
  const int ln   = lane & 15;
  const int bh   = blockIdx.y;   // b*8 + h
  const int b    = bh >> 3;
  const int h    = bh & 7;
  const int q0   = blockIdx.x * 64;

  const float* Qg = qh + ((size_t)bh * 1024 + q0) * 64;
#pragma unroll
  for (int t = 0; t < 8; ++t) {                 // Q tile: 64x64 = 1024 float4
    int i = tid + t * 128;
    int r = i >> 4, c4 = (i & 15) << 2;
    *(float4*)&Qs[r][c4] = *(const float4*)(Qg + (size_t)r * 64 + c4);
  }

  v8f o[4] = {v8f{}, v8f{}, v8f{}, v8f{}};      // 16 rows x 64 dv
  float mrow[8], lrow[8];
#pragma unroll
  for (int vr = 0; vr < 8; ++vr) { mrow[vr] = NEG_INF; lrow[vr] = 0.f; }

  float* aout = attn_out + (size_t)bh * 1024 * 1024;
  const unsigned ks_off = (unsigned)(uintptr_t)&Ks[0][0];
  const unsigned vs_off = (unsigned)(uintptr_t)&Vs[0][0];

  for (int kt = 0; kt < 1024; kt += 32) {
    __syncthreads();                            // Ks/Vs/Ps consumers done
    if (wid == 0) {                             // one wave drives the TDM
      tdm_load_tile64_f32(kh + ((size_t)bh * 1024 + kt) * 64, ks_off, 32u);
      tdm_load_tile64_f32(vh + ((size_t)bh * 1024 + kt) * 64, vs_off, 32u);
      __builtin_amdgcn_s_wait_tensorcnt(0);     // DMA complete before barrier
    }
    __syncthreads();

    // ---- S = Q K^T over this wave's 16 rows x 32 keys
    v8f s[2] = {v8f{}, v8f{}};
#pragma unroll
    for (int kk = 0; kk < 64; kk += 4) {
      const int ka = kk + hi * 2;
      v2f a; a.x = Qs[wid * 16 + ln][ka]; a.y = Qs[wid * 16 + ln][ka + 1];
#pragma unroll
      for (int t = 0; t < 2; ++t) {             // B = K^T: contiguous K-row reads
        v2f bb; bb.x = Ks[t * 16 + ln][ka]; bb.y = Ks[t * 16 + ln][ka + 1];
        s[t] = wmma4(a, bb, s[t]);
      }
    }

    // ---- scale, emit unnorm_attn (unmasked), then mask with -inf
#pragma unroll
    for (int t = 0; t < 2; ++t) {
#pragma unroll
      for (int vr = 0; vr < 8; ++vr) {
        const int qg = q0 + wid * 16 + vr + hi * 8;
        const int kc = kt + t * 16 + ln;
        const float v = s[t][vr] * 0.125f;      // 1/sqrt(64)
        aout[(size_t)qg * 1024 + kc] = v;
        const int mv = mask[((size_t)b * 1024 + qg) * 1024 + kc];
        s[t][vr] = mv ? v : NEG_INF;
      }
    }

    // ---- online softmax (row lives in one 16-lane group -> shfl_xor 1,2,4,8)
#pragma unroll
    for (int vr = 0; vr < 8; ++vr) {
      float tmax = fmaxf(s[0][vr], s[1][vr]);
      tmax = fmaxf(tmax, __shfl_xor(tmax, 1, 32));
      tmax = fmaxf(tmax, __shfl_xor(tmax, 2, 32));
      tmax = fmaxf(tmax, __shfl_xor(tmax, 4, 32));
      tmax = fmaxf(tmax, __shfl_xor(tmax, 8, 32));
      const float mnew = fmaxf(mrow[vr], tmax);
      const float sc   = (mrow[vr] > NEG_INF) ? __expf(mrow[vr] - mnew) : 0.0f;
      float psum = 0.f;
#pragma unroll
      for (int t = 0; t < 2; ++t) {
        const float p = (mnew > NEG_INF) ? __expf(s[t][vr] - mnew) : 0.0f;
        s[t][vr] = p;
        psum += p;
      }
      psum += __shfl_xor(psum, 1, 32);
      psum += __shfl_xor(psum, 2, 32);
      psum += __shfl_xor(psum, 4, 32);
      psum += __shfl_xor(psum, 8, 32);
      lrow[vr] = lrow[vr] * sc + psum;
      mrow[vr] = mnew;
#pragma unroll
      for (int t = 0; t < 4; ++t) o[t][vr] *= sc;
    }

    // ---- relay P through LDS (C-fragment -> A-fragment layout change)
#pragma unroll
    for (int t = 0; t < 2; ++t)
#pragma unroll
      for (int vr = 0; vr < 8; ++vr)
        Ps[wid * 16 + vr + hi * 8][t * 16 + ln] = s[t][vr];
    __syncthreads();

    // ---- O += P(16x32) @ V(32x64)
#pragma unroll
    for (int kk = 0; kk < 32; kk += 4) {
      const int ka = kk + hi * 2;
      v2f a; a.x = Ps[wid * 16 + ln][ka]; a.y = Ps[wid * 16 + ln][ka + 1];
#pragma unroll
      for (int t = 0; t < 4; ++t) {
        v2f bb; bb.x = Vs[ka][t * 16 + ln]; bb.y = Vs[ka + 1][t * 16 + ln];
        o[t] = wmma4(a, bb, o[t]);
      }
    }
  }

  // ---- normalize (all-masked rows: l==0 -> exact 0) and write concat-heads
#pragma unroll
  for (int vr = 0; vr < 8; ++vr) {
    const int qg = q0 + wid * 16 + vr + hi * 8;
    const float inv = (lrow[vr] > 0.f) ? (1.0f / lrow[vr]) : 0.0f;
#pragma unroll
    for (int t = 0; t < 4; ++t)
      o_cat[((size_t)(b * 1024 + qg)) * 512 + h * 64 + t * 16 + ln] = o[t][vr] * inv;
  }
}

// ---------------------------------------------------------------------------
extern "C" void kernel_launch(void* const* d_in, const int* in_sizes, int n_in,
                              void* d_out, int out_size, void* d_ws, size_t ws_size,
                              hipStream_t stream) {
  const float* queries = (const float*)d_in[0];
  const float* keys    = (const float*)d_in[1];
  const float* values  = (const float*)d_in[2];
  const int*   maskp   = (const int*)d_in[3];
  const float* Wq = (const float*)d_in[4];
  const float* bq = (const float*)d_in[5];
  const float* Wk = (const float*)d_in[6];
  const float* bk = (const float*)d_in[7];
  const float* Wv = (const float*)d_in[8];
  const float* bv = (const float*)d_in[9];
  const float* Wo = (const float*)d_in[10];
  const float* bo = (const float*)d_in[11];

  float* out  = (float*)d_out;                       // [8,1024,512]
  float* attn = out + (size_t)8 * 1024 * 512;        // [8,8,1024,1024]

  // workspace: q,k,v head-major (16 MB each) + o_cat (16 MB) = 64 MB
  float* qh   = (float*)d_ws;
  float* khp  = qh  + (size_t)4 * 1024 * 1024;
  float* vhp  = khp + (size_t)4 * 1024 * 1024;
  float* ocat = vhp + (size_t)4 * 1024 * 1024;

  const dim3 gG(8, 128), bG(256);
  hipLaunchKernelGGL(gemm512_wmma_kernel, gG, bG, 0, stream, queries, Wq, bq, qh, 1);
  hipLaunchKernelGGL(gemm512_wmma_kernel, gG, bG, 0, stream, keys,    Wk, bk, khp, 1);
  hipLaunchKernelGGL(gemm512_wmma_kernel, gG, bG, 0, stream, values,  Wv, bv, vhp, 1);
  hipLaunchKernelGGL(attn_flash_wmma_kernel, dim3(16, 64), dim3(128), 0, stream,
                     qh, khp, vhp, maskp, attn, ocat);
  hipLaunchKernelGGL(gemm512_wmma_kernel, gG, bG, 0, stream, ocat, Wo, bo, out, 0);
}